// Interface_72859825209742
// MI455X (gfx1250) — compile-verified
//
#include <hip/hip_runtime.h>
#include <hip/hip_bf16.h>
#include <math.h>

typedef __attribute__((ext_vector_type(2))) float v2f;
typedef __attribute__((ext_vector_type(8))) float v8f;

#define DEVINL __device__ __forceinline__

// ---- problem constants ----
constexpr int B_ = 64, HID_ = 1024, A_ = 8192, C_ = 128, NH_ = 2;
constexpr int P_READ = 134, P_WRITE = 390;
constexpr int P_TOT = NH_ * P_READ + P_WRITE;   // 658
constexpr int P_PAD = 672;                      // 42 * 16
constexpr float EPSF = 1e-12f;

// ---- workspace layout (float offsets) ----
constexpr size_t OFF_WCAT = 0;
constexpr size_t OFF_BIAS = OFF_WCAT + (size_t)P_PAD * HID_;
constexpr size_t OFF_PROJ = OFF_BIAS + P_PAD;
constexpr size_t OFF_QN   = OFF_PROJ + (size_t)B_ * P_PAD;
constexpr size_t OFF_SCAL = OFF_QN + 3ull * B_ * C_;
constexpr size_t OFF_ER   = OFF_SCAL + 3ull * B_ * 8;
constexpr size_t OFF_AD   = OFF_ER + (size_t)B_ * C_;
constexpr size_t OFF_ESIM = OFF_AD + (size_t)B_ * C_;
constexpr size_t OFF_SHRP = OFF_ESIM + 3ull * B_ * A_;
constexpr size_t OFF_SUME = OFF_SHRP + 3ull * B_ * A_;
constexpr size_t OFF_SUMS = OFF_SUME + 3ull * B_;

// ---- d_out layout (return-order concat) ----
constexpr size_t RV_OFF   = 0;                                   // read_vecs [NH,B,C]
constexpr size_t MEM_OFF  = RV_OFF + (size_t)NH_ * B_ * C_;      // memory   [B,A,C]
constexpr size_t RATT_OFF = MEM_OFF + (size_t)B_ * A_ * C_;      // read_att [NH,B,A]
constexpr size_t WATT_OFF = RATT_OFF + (size_t)NH_ * B_ * A_;    // w_att    [B,A]

DEVINL float sigm(float x)  { return 1.f / (1.f + __expf(-x)); }
DEVINL float splus(float x) { return x > 20.f ? x : log1pf(__expf(x)); }
DEVINL float4 ld4(const float* p) { return *reinterpret_cast<const float4*>(p); }
DEVINL void  st4(float* p, float4 v) { *reinterpret_cast<float4*>(p) = v; }
DEVINL float dot4(float4 a, float4 b) {
  return fmaf(a.x, b.x, fmaf(a.y, b.y, fmaf(a.z, b.z, a.w * b.w)));
}
DEVINL float wred(float v) {
  #pragma unroll
  for (int o = 16; o > 0; o >>= 1) v += __shfl_xor(v, o, 32);
  return v;
}

// ---------------------------------------------------------------------------
// K0: pack W_read + W_write into one [P_PAD, HID] matrix (+bias), zero-pad
// ---------------------------------------------------------------------------
__global__ void k_pack(const float* __restrict__ Wr, const float* __restrict__ br,
                       const float* __restrict__ Ww, const float* __restrict__ bw,
                       float* __restrict__ Wcat, float* __restrict__ bias) {
  const int p = blockIdx.x;             // 0 .. P_PAD-1
  const float* src = nullptr;
  float bv = 0.f;
  if (p < NH_ * P_READ)      { src = Wr + (size_t)p * HID_;                 bv = br[p]; }
  else if (p < P_TOT)        { src = Ww + (size_t)(p - NH_ * P_READ) * HID_; bv = bw[p - NH_ * P_READ]; }
  for (int k = threadIdx.x; k < HID_; k += blockDim.x)
    Wcat[(size_t)p * HID_ + k] = src ? src[k] : 0.f;
  if (threadIdx.x == 0) bias[p] = bv;
}

// ---------------------------------------------------------------------------
// K1: zero the accumulators (read_vecs region, sumE, sumS)
// ---------------------------------------------------------------------------
__global__ void k_init(float* __restrict__ out, float* __restrict__ sumE,
                       float* __restrict__ sumS) {
  const int i = blockIdx.x * blockDim.x + threadIdx.x;
  const int NRV = NH_ * B_ * C_;        // 16384
  if (i < NRV)                 out[RV_OFF + i] = 0.f;
  else if (i < NRV + 3 * B_)   sumE[i - NRV] = 0.f;
  else if (i < NRV + 6 * B_)   sumS[i - NRV - 3 * B_] = 0.f;
}

// ---------------------------------------------------------------------------
// K2: projection GEMM with V_WMMA_F32_16X16X4_F32 (exact fp32)
//     D[16b x 16p] += A[16b x 4k] * B[4k x 16p], K = 1024
// one wave (32 threads) per 16x16 tile; grid = (P_PAD/16, B/16)
// ---------------------------------------------------------------------------
__global__ void __launch_bounds__(32)
k_gemm_wmma(const float* __restrict__ ctrl, const float* __restrict__ Wcat,
            const float* __restrict__ bias, float* __restrict__ proj) {
  const int p0 = blockIdx.x * 16;
  const int b0 = blockIdx.y * 16;
  const int lane = threadIdx.x;
  const int mn = lane & 15;              // M (for A) / N (for B,D)
  const int kb = (lane >> 4) << 1;       // lanes 0-15 -> K{0,1}; 16-31 -> K{2,3}

  // 8-byte aligned v2f loads (kb even, k0 multiple of 4)
  const v2f* a2 = reinterpret_cast<const v2f*>(ctrl + (size_t)(b0 + mn) * HID_ + kb);
  const v2f* b2 = reinterpret_cast<const v2f*>(Wcat + (size_t)(p0 + mn) * HID_ + kb);

  v8f acc = {};
  #pragma unroll 8
  for (int k0 = 0; k0 < HID_; k0 += 4) {
    v2f av = a2[k0 >> 1];
    v2f bv = b2[k0 >> 1];
    acc = __builtin_amdgcn_wmma_f32_16x16x4_f32(
        /*neg_a=*/false, av, /*neg_b=*/false, bv,
        /*c_mod=*/(short)0, acc, /*reuse_a=*/false, /*reuse_b=*/false);
  }

  const int p = p0 + mn;
  if (p < P_TOT) {
    const float bv = bias[p];
    const int mbase = (lane >> 4) * 8;   // D: vgpr r -> M = r (+8 for lanes 16-31)
    #pragma unroll
    for (int r = 0; r < 8; ++r)
      proj[(size_t)(b0 + mbase + r) * P_PAD + p] = acc[r] + bv;
  }
}

// ---------------------------------------------------------------------------
// K3: activations: sigmoid(q)+l2norm, per-head scalars, erase/add
// grid = (3, B), block = 128
// ---------------------------------------------------------------------------
__global__ void __launch_bounds__(128)
k_act(const float* __restrict__ proj, float* __restrict__ qn,
      float* __restrict__ scal, float* __restrict__ er, float* __restrict__ ad) {
  const int h = blockIdx.x, b = blockIdx.y, t = threadIdx.x;
  const int base = (h < 2) ? h * P_READ : NH_ * P_READ;
  const float* pr = proj + (size_t)b * P_PAD + base;

  const float qv = sigm(pr[t]);
  float ss = wred(qv * qv);
  __shared__ float red[4];
  if ((t & 31) == 0) red[t >> 5] = ss;
  __syncthreads();
  const float nrm = sqrtf(red[0] + red[1] + red[2] + red[3]);
  qn[((size_t)h * B_ + b) * C_ + t] = qv / fmaxf(nrm, EPSF);

  if (t == 0) {
    float* sc = scal + ((size_t)h * B_ + b) * 8;
    sc[0] = splus(pr[C_]) + 1.f;                 // beta
    sc[1] = sigm(pr[C_ + 1]);                    // gate
    float s0 = pr[C_ + 2], s1 = pr[C_ + 3], s2 = pr[C_ + 4];
    float mx = fmaxf(s0, fmaxf(s1, s2));
    float e0 = __expf(s0 - mx), e1 = __expf(s1 - mx), e2 = __expf(s2 - mx);
    float es = e0 + e1 + e2;
    sc[2] = e0 / es; sc[3] = e1 / es; sc[4] = e2 / es;
    sc[5] = splus(pr[C_ + 5]) + 1.f;             // gamma
  }
  if (h == 2) {
    er[(size_t)b * C_ + t] = sigm(pr[C_ + 6 + t]);
    ad[(size_t)b * C_ + t] = sigm(pr[2 * C_ + 6 + t]);
  }
}

// ---------------------------------------------------------------------------
// K4: BIG PASS 1 — per memory row: norm + 3 cosine dots -> exp(beta*(sim-1))
// grid = (A/64, B), block = 256 (8 waves, wave-per-row, 8 rows each)
// ---------------------------------------------------------------------------
__global__ void __launch_bounds__(256)
k_sim(const float* __restrict__ mem, const float* __restrict__ qn,
      const float* __restrict__ scal, float* __restrict__ esim,
      float* __restrict__ sumE) {
  const int b = blockIdx.y;
  const int lane = threadIdx.x & 31, wave = threadIdx.x >> 5;
  const int c4 = lane * 4;

  const float4 q0 = ld4(qn + ((size_t)0 * B_ + b) * C_ + c4);
  const float4 q1 = ld4(qn + ((size_t)1 * B_ + b) * C_ + c4);
  const float4 q2 = ld4(qn + ((size_t)2 * B_ + b) * C_ + c4);
  const float bt0 = scal[((size_t)0 * B_ + b) * 8];
  const float bt1 = scal[((size_t)1 * B_ + b) * 8];
  const float bt2 = scal[((size_t)2 * B_ + b) * 8];

  __shared__ float lsum[3];
  if (threadIdx.x < 3) lsum[threadIdx.x] = 0.f;
  __syncthreads();

  float p0 = 0.f, p1 = 0.f, p2 = 0.f;
  #pragma unroll 2
  for (int i = 0; i < 8; ++i) {
    const int a = blockIdx.x * 64 + i * 8 + wave;
    const float4 m = ld4(mem + ((size_t)b * A_ + a) * C_ + c4);
    float ss = wred(dot4(m, m));
    float d0 = wred(dot4(m, q0));
    float d1 = wred(dot4(m, q1));
    float d2 = wred(dot4(m, q2));
    if (lane == 0) {
      const float inv = 1.f / fmaxf(sqrtf(ss), EPSF);
      const float e0 = __expf(bt0 * (d0 * inv - 1.f));
      const float e1 = __expf(bt1 * (d1 * inv - 1.f));
      const float e2 = __expf(bt2 * (d2 * inv - 1.f));
      esim[((size_t)0 * B_ + b) * A_ + a] = e0;
      esim[((size_t)1 * B_ + b) * A_ + a] = e1;
      esim[((size_t)2 * B_ + b) * A_ + a] = e2;
      p0 += e0; p1 += e1; p2 += e2;
    }
  }
  if (lane == 0) {
    atomicAdd(&lsum[0], p0); atomicAdd(&lsum[1], p1); atomicAdd(&lsum[2], p2);
  }
  __syncthreads();
  if (threadIdx.x < 3) atomicAdd(&sumE[(size_t)threadIdx.x * B_ + b], lsum[threadIdx.x]);
}

// ---------------------------------------------------------------------------
// K5: softmax-normalize, gate, circular shift, sharpen; accumulate sums
// grid = (A/256, 3*B), block = 256
// ---------------------------------------------------------------------------
__global__ void __launch_bounds__(256)
k_sharp(const float* __restrict__ esim, const float* __restrict__ sumE,
        const float* __restrict__ scal, const float* __restrict__ prev_r,
        const float* __restrict__ prev_w, float* __restrict__ sharp,
        float* __restrict__ sumS) {
  const int hb = blockIdx.y;
  const int h = hb / B_, b = hb % B_;
  const int a = blockIdx.x * 256 + threadIdx.x;
  const float* sc = scal + (size_t)hb * 8;
  const float gate = sc[1], s0 = sc[2], s1 = sc[3], s2 = sc[4], gamma = sc[5];
  const float invE = 1.f / sumE[hb];
  const float* es = esim + (size_t)hb * A_;
  const float* prev = (h < 2) ? (prev_r + ((size_t)h * B_ + b) * A_)
                              : (prev_w + (size_t)b * A_);
  const int am1 = (a + A_ - 1) & (A_ - 1);
  const int ap1 = (a + 1) & (A_ - 1);
  const float gm1 = gate * (es[am1] * invE) + (1.f - gate) * prev[am1];
  const float g0  = gate * (es[a]   * invE) + (1.f - gate) * prev[a];
  const float gp1 = gate * (es[ap1] * invE) + (1.f - gate) * prev[ap1];
  const float sh = fmaf(s0, gm1, fmaf(s1, g0, s2 * gp1));
  const float sp = __expf(gamma * __logf(sh + EPSF));   // (sh+eps)^gamma, sh>=0
  sharp[(size_t)hb * A_ + a] = sp;

  float v = wred(sp);
  __shared__ float red[8];
  if ((threadIdx.x & 31) == 0) red[threadIdx.x >> 5] = v;
  __syncthreads();
  if (threadIdx.x == 0) {
    float t = 0.f;
    #pragma unroll
    for (int i = 0; i < 8; ++i) t += red[i];
    atomicAdd(&sumS[hb], t);
  }
}

// ---------------------------------------------------------------------------
// K6: normalize sharpened attention -> d_out (read_att / w_att regions)
// ---------------------------------------------------------------------------
__global__ void __launch_bounds__(256)
k_att(const float* __restrict__ sharp, const float* __restrict__ sumS,
      float* __restrict__ out) {
  const int hb = blockIdx.y;
  const int h = hb / B_, b = hb % B_;
  const int a = blockIdx.x * 256 + threadIdx.x;
  const float v = sharp[(size_t)hb * A_ + a] / sumS[hb];
  if (h < 2) out[RATT_OFF + ((size_t)h * B_ + b) * A_ + a] = v;
  else       out[WATT_OFF + (size_t)b * A_ + a] = v;
}

// ---------------------------------------------------------------------------
// K7: BIG PASS 2 — fused read_vecs contraction + erase/add memory update
// grid = (A/256, B), block = 256 (8 waves, wave-per-row, 32 rows each)
// ---------------------------------------------------------------------------
__global__ void __launch_bounds__(256)
k_final(const float* __restrict__ mem, const float* __restrict__ er,
        const float* __restrict__ ad, float* __restrict__ out) {
  const int b = blockIdx.y;
  const int lane = threadIdx.x & 31, wave = threadIdx.x >> 5;
  const int c4 = lane * 4;

  const float4 e4 = ld4(er + (size_t)b * C_ + c4);
  const float4 a4 = ld4(ad + (size_t)b * C_ + c4);
  const float* ra0 = out + RATT_OFF + (size_t)b * A_;
  const float* ra1 = out + RATT_OFF + ((size_t)B_ + b) * A_;
  const float* wa  = out + WATT_OFF + (size_t)b * A_;
  float* memout = out + MEM_OFF;

  float4 acc0 = {0.f, 0.f, 0.f, 0.f}, acc1 = {0.f, 0.f, 0.f, 0.f};
  for (int i = 0; i < 32; ++i) {
    const int a = blockIdx.x * 256 + i * 8 + wave;
    const size_t ro = ((size_t)b * A_ + a) * C_ + c4;
    const float4 m = ld4(mem + ro);
    const float t0 = ra0[a], t1 = ra1[a], w = wa[a];
    acc0.x = fmaf(t0, m.x, acc0.x); acc0.y = fmaf(t0, m.y, acc0.y);
    acc0.z = fmaf(t0, m.z, acc0.z); acc0.w = fmaf(t0, m.w, acc0.w);
    acc1.x = fmaf(t1, m.x, acc1.x); acc1.y = fmaf(t1, m.y, acc1.y);
    acc1.z = fmaf(t1, m.z, acc1.z); acc1.w = fmaf(t1, m.w, acc1.w);
    float4 o;
    o.x = fmaf(m.x, 1.f - w * e4.x, w * a4.x);
    o.y = fmaf(m.y, 1.f - w * e4.y, w * a4.y);
    o.z = fmaf(m.z, 1.f - w * e4.z, w * a4.z);
    o.w = fmaf(m.w, 1.f - w * e4.w, w * a4.w);
    st4(memout + ro, o);
  }

  // reduce per-wave partials across the block, then atomicAdd to read_vecs
  __shared__ float red[8][2][C_];
  red[wave][0][c4 + 0] = acc0.x; red[wave][0][c4 + 1] = acc0.y;
  red[wave][0][c4 + 2] = acc0.z; red[wave][0][c4 + 3] = acc0.w;
  red[wave][1][c4 + 0] = acc1.x; red[wave][1][c4 + 1] = acc1.y;
  red[wave][1][c4 + 2] = acc1.z; red[wave][1][c4 + 3] = acc1.w;
  __syncthreads();
  const int head = threadIdx.x >> 7;        // 0..1
  const int c = threadIdx.x & (C_ - 1);     // 0..127
  float t = 0.f;
  #pragma unroll
  for (int w8 = 0; w8 < 8; ++w8) t += red[w8][head][c];
  atomicAdd(&out[RV_OFF + ((size_t)head * B_ + b) * C_ + c], t);
}

// ---------------------------------------------------------------------------
extern "C" void kernel_launch(void* const* d_in, const int* in_sizes, int n_in,
                              void* d_out, int out_size, void* d_ws, size_t ws_size,
                              hipStream_t stream) {
  const float* ctrl   = (const float*)d_in[0];
  const float* memory = (const float*)d_in[1];
  const float* pratt  = (const float*)d_in[2];
  const float* pwatt  = (const float*)d_in[3];
  const float* Wr     = (const float*)d_in[4];
  const float* br     = (const float*)d_in[5];
  const float* Ww     = (const float*)d_in[6];
  const float* bw     = (const float*)d_in[7];
  float* out = (float*)d_out;
  float* ws  = (float*)d_ws;

  float* Wcat = ws + OFF_WCAT;
  float* bias = ws + OFF_BIAS;
  float* proj = ws + OFF_PROJ;
  float* qn   = ws + OFF_QN;
  float* scal = ws + OFF_SCAL;
  float* er   = ws + OFF_ER;
  float* ad   = ws + OFF_AD;
  float* esim = ws + OFF_ESIM;
  float* shrp = ws + OFF_SHRP;
  float* sumE = ws + OFF_SUME;
  float* sumS = ws + OFF_SUMS;

  k_pack<<<dim3(P_PAD), 256, 0, stream>>>(Wr, br, Ww, bw, Wcat, bias);
  k_init<<<dim3((NH_ * B_ * C_ + 6 * B_ + 255) / 256), 256, 0, stream>>>(out, sumE, sumS);
  k_gemm_wmma<<<dim3(P_PAD / 16, B_ / 16), 32, 0, stream>>>(ctrl, Wcat, bias, proj);
  k_act<<<dim3(3, B_), C_, 0, stream>>>(proj, qn, scal, er, ad);
  k_sim<<<dim3(A_ / 64, B_), 256, 0, stream>>>(memory, qn, scal, esim, sumE);
  k_sharp<<<dim3(A_ / 256, 3 * B_), 256, 0, stream>>>(esim, sumE, scal, pratt, pwatt, shrp, sumS);
  k_att<<<dim3(A_ / 256, 3 * B_), 256, 0, stream>>>(shrp, sumS, out);
  k_final<<<dim3(A_ / 256, B_), 256, 0, stream>>>(memory, er, ad, out);
}